// ChunkAttentionWithRelativeRightContext_7627861917918
// MI455X (gfx1250) — compile-verified
//
#include <hip/hip_runtime.h>

typedef float v2f __attribute__((ext_vector_type(2)));
typedef float v8f __attribute__((ext_vector_type(8)));

#define B_    4
#define T1_   512
#define T2_   832
#define NPOS_ 1343
#define NF_   1024
#define H_    16
#define DK_   64
#define WIN_  384   // keys per chunk window
#define QB_   32    // queries per workgroup (half a chunk) -> 48KB LDS scores

static __device__ __forceinline__ v8f wmma4(v2f a, v2f b, v8f c) {
  // D = A(16x4,f32) * B(4x16,f32) + C(16x16,f32)
  return __builtin_amdgcn_wmma_f32_16x16x4_f32(false, a, false, b, (short)0, c,
                                               false, false);
}

// Y(M x N) = X(M x K) @ W(N x K)^T + bias, 32x32 output tile per wave.
// mode 0: Y row-major (M x N)
// mode 1: Y is K-half of attention cache  [b][h][t][0:64]   (t2 = rows per batch)
// mode 2: Y is V-half of attention cache  [b][h][t][64:128]
__global__ __launch_bounds__(256) void gemm_xwT_kernel(
    const float* __restrict__ X, const float* __restrict__ W,
    const float* __restrict__ bias, float* __restrict__ Y,
    int M, int K, int N, int mode, int t2) {
  const int ntiles = N >> 5;                 // 32-wide tiles (N multiple of 32)
  const int mtiles = (M + 31) >> 5;
  const int tile = blockIdx.x * 8 + (threadIdx.x >> 5);   // wave-uniform
  if (tile >= mtiles * ntiles) return;
  const int mt = tile / ntiles, nt = tile - mt * ntiles;
  const int lane = threadIdx.x & 31;
  const int ln = lane & 15, half = lane >> 4;

  int rowA0 = mt * 32 + ln;      if (rowA0 >= M) rowA0 = M - 1;  // clamp (P: M=1343)
  int rowA1 = mt * 32 + 16 + ln; if (rowA1 >= M) rowA1 = M - 1;
  const int colB0 = nt * 32 + ln;
  const int colB1 = nt * 32 + 16 + ln;
  const float* ap0 = X + (size_t)rowA0 * K + half * 2;
  const float* ap1 = X + (size_t)rowA1 * K + half * 2;
  const float* bp0 = W + (size_t)colB0 * K + half * 2;
  const float* bp1 = W + (size_t)colB1 * K + half * 2;

  v8f acc00 = {}, acc01 = {}, acc10 = {}, acc11 = {};
#pragma unroll 4
  for (int k0 = 0; k0 < K; k0 += 4) {
    v2f a0 = *(const v2f*)(ap0 + k0);
    v2f a1 = *(const v2f*)(ap1 + k0);
    v2f b0 = *(const v2f*)(bp0 + k0);
    v2f b1 = *(const v2f*)(bp1 + k0);
    acc00 = wmma4(a0, b0, acc00);
    acc01 = wmma4(a0, b1, acc01);
    acc10 = wmma4(a1, b0, acc10);
    acc11 = wmma4(a1, b1, acc11);
  }

  const float cb0 = bias ? bias[colB0] : 0.0f;
  const float cb1 = bias ? bias[colB1] : 0.0f;
#pragma unroll
  for (int ti = 0; ti < 2; ++ti) {
#pragma unroll
    for (int tj = 0; tj < 2; ++tj) {
      const v8f& acc = ti ? (tj ? acc11 : acc10) : (tj ? acc01 : acc00);
      const int col = nt * 32 + tj * 16 + ln;
      const float cb = tj ? cb1 : cb0;
#pragma unroll
      for (int i = 0; i < 8; ++i) {
        const int row = mt * 32 + ti * 16 + i + half * 8;
        if (row >= M) continue;
        const float v = acc[i] + cb;
        size_t idx;
        if (mode == 0) {
          idx = (size_t)row * N + col;
        } else {
          const int b = row / t2, t = row - b * t2;
          const int h = col >> 6, d = col & 63;
          idx = (((size_t)b * H_ + h) * t2 + t) * 128 + d + (mode == 2 ? 64 : 0);
        }
        Y[idx] = v;
      }
    }
  }
}

// Fused banded attention: one WG = (b, h, 32 queries); keys [64*chunk, +384).
// cache layout: [b][h][t2][0:64]=K, [64:128]=V (written by gemm modes 1/2).
__global__ __launch_bounds__(256) void attn_kernel(
    const float* __restrict__ qws, const float* __restrict__ pws,
    const float* __restrict__ cache,
    const float* __restrict__ pbu, const float* __restrict__ pbv,
    float* __restrict__ xws) {
  __shared__ float sc[QB_ * WIN_];   // 48 KB

  const int bid = blockIdx.x;
  const int qb = bid & 15;           // 16 q-blocks of 32
  const int h  = (bid >> 4) & 15;
  const int b  = bid >> 8;
  const int q0 = qb * QB_;           // query base within batch
  const int chunk = q0 >> 6;
  const int j0 = chunk * 64;         // key window start
  const int qi_base = q0 & 63;       // 0 or 32 (local-in-chunk)

  const int wave = threadIdx.x >> 5;
  const int lane = threadIdx.x & 31;
  const int ln = lane & 15, half = lane >> 4;

  // ---- AC = (q + pos_bias_u) @ K^T  -> LDS ----
  for (int t = wave; t < 2 * 24; t += 8) {
    const int qt = t / 24, jt = t - qt * 24;
    const int rowA = q0 + qt * 16 + ln;
    const float* ap = qws + (size_t)(b * T1_ + rowA) * NF_ + h * DK_ + half * 2;
    const float* up = pbu + h * DK_ + half * 2;
    const int j = j0 + jt * 16 + ln;
    const float* bp = cache + (((size_t)b * H_ + h) * T2_ + j) * 128 + half * 2;
    v8f acc = {};
#pragma unroll
    for (int k0 = 0; k0 < DK_; k0 += 4) {
      v2f a = *(const v2f*)(ap + k0);
      v2f u = *(const v2f*)(up + k0);
      a = a + u;
      v2f bb = *(const v2f*)(bp + k0);
      acc = wmma4(a, bb, acc);
    }
#pragma unroll
    for (int i = 0; i < 8; ++i)
      sc[(qt * 16 + i + half * 8) * WIN_ + jt * 16 + ln] = acc[i];
  }
  __syncthreads();

  // ---- BD = (q + pos_bias_v) @ P^T on the needed band, shifted add ----
  // ji = ncol - 511 + qi ; per 16-row tile the needed ncol span is 399 wide.
  for (int t = wave; t < 2 * 25; t += 8) {
    const int qt = t / 25, nt = t - qt * 25;
    const int qi0 = qi_base + qt * 16;
    const int n0 = (T1_ - 16) - qi0;             // 496 - qi0
    const int rowA = q0 + qt * 16 + ln;
    const float* ap = qws + (size_t)(b * T1_ + rowA) * NF_ + h * DK_ + half * 2;
    const float* vp = pbv + h * DK_ + half * 2;
    const int ncol = n0 + nt * 16 + ln;          // in [448, 895] subset of NPOS
    const float* bp = pws + (size_t)ncol * NF_ + h * DK_ + half * 2;
    v8f acc = {};
#pragma unroll
    for (int k0 = 0; k0 < DK_; k0 += 4) {
      v2f a = *(const v2f*)(ap + k0);
      v2f u = *(const v2f*)(vp + k0);
      a = a + u;
      v2f bb = *(const v2f*)(bp + k0);
      acc = wmma4(a, bb, acc);
    }
#pragma unroll
    for (int i = 0; i < 8; ++i) {
      const int mloc = i + half * 8;
      const int qi = qi0 + mloc;                 // local-in-chunk row
      const int ji = ncol - (T1_ - 1) + qi;      // rel-shift
      if (ji >= 0 && ji < WIN_)
        sc[(qt * 16 + mloc) * WIN_ + ji] += acc[i];   // unique (row,ji) per tile
    }
  }
  __syncthreads();

  // ---- softmax over the 384 in-window (all allowed) positions ----
  if (threadIdx.x < QB_) {
    float* r = sc + threadIdx.x * WIN_;
    const float scale = 0.125f;                  // 1/sqrt(DK)
    float mx = -3.402823e38f;
    for (int j = 0; j < WIN_; ++j) { float v = r[j] * scale; r[j] = v; mx = fmaxf(mx, v); }
    float s = 0.0f;
    for (int j = 0; j < WIN_; ++j) { float e = __expf(r[j] - mx); r[j] = e; s += e; }
    const float inv = 1.0f / s;
    for (int j = 0; j < WIN_; ++j) r[j] *= inv;
  }
  __syncthreads();

  // ---- attn(32x384) @ V(384x64) -> x[b][q][h*64+d] ----
  {
    const int qt = wave >> 2, nt = wave & 3;     // 8 waves = 2x4 tiles
    const int n = nt * 16 + ln;
    const float* ap = (const float*)sc + (qt * 16 + ln) * WIN_ + half * 2;
    const float* vbase = cache + (((size_t)b * H_ + h) * T2_ + j0) * 128 + 64 + n;
    v8f acc = {};
#pragma unroll 8
    for (int k0 = 0; k0 < WIN_; k0 += 4) {
      v2f a = *(const v2f*)(ap + k0);
      const int k = k0 + half * 2;
      v2f bb;
      bb.x = vbase[(size_t)k * 128];
      bb.y = vbase[(size_t)(k + 1) * 128];
      acc = wmma4(a, bb, acc);
    }
#pragma unroll
    for (int i = 0; i < 8; ++i) {
      const int rowg = q0 + qt * 16 + i + half * 8;
      xws[(size_t)(b * T1_ + rowg) * NF_ + h * DK_ + nt * 16 + ln] = acc[i];
    }
  }
}

extern "C" void kernel_launch(void* const* d_in, const int* in_sizes, int n_in,
                              void* d_out, int out_size, void* d_ws, size_t ws_size,
                              hipStream_t stream) {
  const float* query = (const float*)d_in[0];
  const float* key   = (const float*)d_in[1];
  const float* value = (const float*)d_in[2];
  const float* pos   = (const float*)d_in[3];
  const float* Wq    = (const float*)d_in[4];
  const float* bq    = (const float*)d_in[5];
  const float* Wk    = (const float*)d_in[6];
  const float* bk    = (const float*)d_in[7];
  const float* Wv    = (const float*)d_in[8];
  const float* bv    = (const float*)d_in[9];
  const float* Wpos  = (const float*)d_in[10];
  const float* Wout  = (const float*)d_in[11];
  const float* bout  = (const float*)d_in[12];
  const float* pbu   = (const float*)d_in[13];
  const float* pbv   = (const float*)d_in[14];
  (void)in_sizes; (void)n_in; (void)out_size; (void)ws_size;

  float* out   = (float*)d_out;                       // (B,T1,1024)
  float* cache = out + (size_t)B_ * T1_ * NF_;        // (B,H,T2,128) = concat(k,v)

  float* qws = (float*)d_ws;                          // (B*T1, 1024)
  float* pws = qws + (size_t)B_ * T1_ * NF_;          // (NPOS, 1024) (padded rows)
  float* xws = pws + (size_t)1344 * NF_;              // (B*T1, 1024)

  auto nb = [](int tiles) { return (tiles + 7) / 8; };
  const int nt32 = NF_ / 32;                          // 32 col-tiles

  // projections (K=1024 -> 256 k-steps, 4 WMMA each, per 32x32 tile)
  gemm_xwT_kernel<<<nb((B_*T1_/32)*nt32), 256, 0, stream>>>(query, Wq, bq, qws, B_*T1_, NF_, NF_, 0, 0);
  gemm_xwT_kernel<<<nb((B_*T2_/32)*nt32), 256, 0, stream>>>(key,   Wk, bk, cache, B_*T2_, NF_, NF_, 1, T2_);
  gemm_xwT_kernel<<<nb((B_*T2_/32)*nt32), 256, 0, stream>>>(value, Wv, bv, cache, B_*T2_, NF_, NF_, 2, T2_);
  gemm_xwT_kernel<<<nb(((NPOS_+31)/32)*nt32), 256, 0, stream>>>(pos, Wpos, nullptr, pws, NPOS_, NF_, NF_, 0, 0);

  // fused banded attention (AC + shifted BD + softmax + PV)
  attn_kernel<<<B_ * H_ * (T1_ / QB_), 256, 0, stream>>>(qws, pws, cache, pbu, pbv, xws);

  // output projection
  gemm_xwT_kernel<<<nb((B_*T1_/32)*nt32), 256, 0, stream>>>(xws, Wout, bout, out, B_*T1_, NF_, NF_, 0, 0);
}